// CrossScan_57251914055738
// MI455X (gfx1250) — compile-verified
//
#include <hip/hip_runtime.h>
#include <hip/hip_bf16.h>

#define Hh   128
#define Ww   128
#define HW   16384
#define CC   96
#define Bb   16
#define TS   32          // tile is TS x TS floats
#define LST  36          // LDS row stride in words (144B, keeps b128 16B-aligned)

typedef float f4  __attribute__((ext_vector_type(4)));
typedef int   v4i __attribute__((ext_vector_type(4)));

typedef __attribute__((address_space(1))) v4i gv4i;   // global (AS1) int4
typedef __attribute__((address_space(3))) v4i lv4i;   // LDS    (AS3) int4

#if defined(__gfx1250__) && __has_builtin(__builtin_amdgcn_global_load_async_to_lds_b128)
#define HAVE_ASYNC_LDS 1
#else
#define HAVE_ASYNC_LDS 0
#endif

__device__ __forceinline__ void wait_asynccnt0() {
#if __has_builtin(__builtin_amdgcn_s_wait_asynccnt)
    __builtin_amdgcn_s_wait_asynccnt(0);
#else
    asm volatile("s_wait_asynccnt 0" ::: "memory");
#endif
}

__global__ __launch_bounds__(256) void cross_scan_kernel(const float* __restrict__ x,
                                                         float* __restrict__ out) {
    __shared__ float tile[TS * LST];

    const int s   = blockIdx.y;            // slice index = b*CC + c
    const int tid = threadIdx.x;           // 0..255
    const int th  = blockIdx.x >> 2;       // tile row  0..3
    const int tw  = blockIdx.x & 3;        // tile col  0..3
    const int h0  = th * TS;
    const int w0  = tw * TS;
    const int r   = tid >> 3;              // 0..31 : row within tile
    const int cg  = tid & 7;               // 0..7  : float4 column group

    const float* src    = x + (size_t)s * HW + (size_t)(h0 + r) * Ww + w0 + 4 * cg;
    float*       ldsdst = &tile[r * LST + 4 * cg];

#if HAVE_ASYNC_LDS
    // gfx1250 async copy: global -> LDS directly, tracked by ASYNCcnt
    __builtin_amdgcn_global_load_async_to_lds_b128(
        (gv4i*)const_cast<float*>(src),
        (lv4i*)ldsdst,
        /*offset=*/0, /*cpol=*/0);
    wait_asynccnt0();
#else
    {
        f4 v = *(const f4*)src;
        *(f4*)ldsdst = v;                  // ds_store_b128
    }
#endif
    __syncthreads();

    // ---- output bases: out[b][dir][c][HW], dir stride = CC*HW ----
    const int    b  = s / CC;
    const int    c  = s - b * CC;
    const size_t o0 = ((size_t)b * 4 * CC + c) * (size_t)HW;
    const size_t ds = (size_t)CC * HW;
    float* out0 = out + o0;
    float* out1 = out + o0 + ds;
    float* out2 = out + o0 + 2 * ds;
    float* out3 = out + o0 + 3 * ds;

    // ---- y0 / y2 : straight read from LDS (ds_load_b128), NT streaming stores ----
    f4 v = *(const f4*)&tile[r * LST + 4 * cg];
    const int i0 = (h0 + r) * Ww + w0 + 4 * cg;
    __builtin_nontemporal_store(v, (f4*)(out0 + i0));
    f4 vr = {v.w, v.z, v.y, v.x};
    __builtin_nontemporal_store(vr, (f4*)(out2 + (HW - 4 - i0)));

    // ---- y1 / y3 : transposed read (4x ds_load_b32 down a column) ----
    const int rw = r;                      // output row index (= original w within tile)
    const int hg = cg;                     // group of 4 original rows
    f4 u;
    u.x = tile[(4 * hg + 0) * LST + rw];
    u.y = tile[(4 * hg + 1) * LST + rw];
    u.z = tile[(4 * hg + 2) * LST + rw];
    u.w = tile[(4 * hg + 3) * LST + rw];
    const int i1 = (w0 + rw) * Hh + h0 + 4 * hg;   // (w*H + h), H==W==128
    __builtin_nontemporal_store(u, (f4*)(out1 + i1));
    f4 ur = {u.w, u.z, u.y, u.x};
    __builtin_nontemporal_store(ur, (f4*)(out3 + (HW - 4 - i1)));
}

extern "C" void kernel_launch(void* const* d_in, const int* in_sizes, int n_in,
                              void* d_out, int out_size, void* d_ws, size_t ws_size,
                              hipStream_t stream) {
    const float* x   = (const float*)d_in[0];
    float*       out = (float*)d_out;
    // 16 tiles of 32x32 per (b,c) slice; 1536 slices
    dim3 grid(16, Bb * CC);
    cross_scan_kernel<<<grid, 256, 0, stream>>>(x, out);
}